// GraphConvLayer_16166256902541
// MI455X (gfx1250) — compile-verified
//
#include <hip/hip_runtime.h>
#include <math.h>

typedef float v2f __attribute__((ext_vector_type(2)));
typedef float v8f __attribute__((ext_vector_type(8)));

#define ROWS 32        // nodes per block (N = 100000 is divisible by 32)
#define KNN 16
#define HD 128
#define MIXK 262       // 2H + 6
#define MIXKP 264      // padded to multiple of 4 (WMMA f32 K step)
#define MSTRIDE 268    // LDS row stride in floats: 268%64=12 banks -> conflict-free, /4 for float4

__global__ __launch_bounds__(256)
void graphconv_wmma_kernel(const float* __restrict__ feat,
                           const float* __restrict__ coords,
                           const int*   __restrict__ knn,
                           const float* __restrict__ W,
                           const float* __restrict__ bias,
                           float* __restrict__ out)
{
    extern __shared__ float smem[];
    float* mixLds = smem;                               // ROWS * MSTRIDE floats
    int*   knnLds = (int*)(smem + ROWS * MSTRIDE);      // ROWS * KNN ints

    const int tid  = threadIdx.x;
    const int base = blockIdx.x * ROWS;

    // ---- stage knn indices in LDS (streamed once -> non-temporal) ----
    for (int i = tid; i < ROWS * KNN; i += 256)
        knnLds[i] = __builtin_nontemporal_load(knn + (size_t)base * KNN + i);
    __syncthreads();

    // ---- phase 1a: copy self feat + neighbor-mean aggregation ----
    {
        const int r    = tid >> 3;     // node within tile, 0..31
        const int sub  = tid & 7;      // 8 threads per node
        const int col0 = sub * 16;     // 16 columns (4 x float4) each
        float* mrow = mixLds + r * MSTRIDE;

        // self features -> mix cols [0,128)
        const float4* sp = (const float4*)(feat + (size_t)(base + r) * HD + col0);
        float4* dst_self = (float4*)(mrow + col0);
        float4* dst_agg  = (float4*)(mrow + HD + col0);
        float ax[16];
        #pragma unroll
        for (int j = 0; j < 4; ++j) {
            float4 v = sp[j];
            dst_self[j] = v;
            ax[4*j+0] = 0.f; ax[4*j+1] = 0.f; ax[4*j+2] = 0.f; ax[4*j+3] = 0.f;
        }
        // gather 16 neighbors from L2-resident feat, accumulate
        #pragma unroll
        for (int k = 0; k < KNN; ++k) {
            const int idx = knnLds[r * KNN + k];
            const float4* np = (const float4*)(feat + (size_t)idx * HD + col0);
            #pragma unroll
            for (int j = 0; j < 4; ++j) {
                float4 v = np[j];
                ax[4*j+0] += v.x; ax[4*j+1] += v.y;
                ax[4*j+2] += v.z; ax[4*j+3] += v.w;
            }
        }
        const float inv = 1.0f / 16.0f;
        #pragma unroll
        for (int j = 0; j < 4; ++j) {
            float4 v;
            v.x = ax[4*j+0]*inv; v.y = ax[4*j+1]*inv;
            v.z = ax[4*j+2]*inv; v.w = ax[4*j+3]*inv;
            dst_agg[j] = v;      // agg -> mix cols [128,256)
        }
    }

    // ---- phase 1b: relative-coordinate mean/std -> mix cols [256,262), pad [262,264) ----
    if (tid < ROWS) {
        const int r = tid;
        const int node = base + r;
        const float cx = coords[(size_t)node*3+0];
        const float cy = coords[(size_t)node*3+1];
        const float cz = coords[(size_t)node*3+2];
        float sx=0.f, sy=0.f, sz=0.f, qx=0.f, qy=0.f, qz=0.f;
        #pragma unroll
        for (int k = 0; k < KNN; ++k) {
            const int idx = knnLds[r * KNN + k];
            float dx = coords[(size_t)idx*3+0] - cx;
            float dy = coords[(size_t)idx*3+1] - cy;
            float dz = coords[(size_t)idx*3+2] - cz;
            sx += dx; sy += dy; sz += dz;
            qx += dx*dx; qy += dy*dy; qz += dz*dz;
        }
        const float inv = 1.0f / 16.0f;
        const float mx = sx*inv, my = sy*inv, mz = sz*inv;
        float vx = qx*inv - mx*mx; vx = vx > 0.f ? vx : 0.f;
        float vy = qy*inv - my*my; vy = vy > 0.f ? vy : 0.f;
        float vz = qz*inv - mz*mz; vz = vz > 0.f ? vz : 0.f;
        float* mrow = mixLds + r * MSTRIDE + 2*HD;
        mrow[0] = mx;        mrow[1] = my;        mrow[2] = mz;
        mrow[3] = sqrtf(vx); mrow[4] = sqrtf(vy); mrow[5] = sqrtf(vz);
        mrow[6] = 0.f;       mrow[7] = 0.f;       // zero pad K = 262,263
    }
    __syncthreads();

    // ---- phase 2: WMMA f32 16x16x4 GEMM: [32 x 264] x [264 x 128] ----
    const int wave    = tid >> 5;        // 0..7 -> 16-column strip of output
    const int lane    = tid & 31;
    const int colbase = wave * 16;
    const int lx      = lane & 15;
    const int hi      = lane >> 4;       // lane half selects K+2 (A & B layouts)

    v8f c0 = {};   // rows 0..15 of tile
    v8f c1 = {};   // rows 16..31 of tile

    const float* mA0 = mixLds + lx * MSTRIDE;
    const float* mA1 = mixLds + (lx + 16) * MSTRIDE;

    // main loop: kk in [0, 260), all K indices valid -> no guards in hot loop
    for (int kk = 0; kk < MIXKP - 4; kk += 4) {
        const int k0 = kk + hi * 2;      // even, <= 258 here
        v2f a0, a1, b;
        a0.x = mA0[k0]; a0.y = mA0[k0 + 1];
        a1.x = mA1[k0]; a1.y = mA1[k0 + 1];
        const float* wp = W + (size_t)k0 * HD + colbase + lx;
        b.x = wp[0];
        b.y = wp[HD];
        c0 = __builtin_amdgcn_wmma_f32_16x16x4_f32(false, a0, false, b,
                                                   (short)0, c0, false, false);
        c1 = __builtin_amdgcn_wmma_f32_16x16x4_f32(false, a1, false, b,
                                                   (short)0, c1, false, false);
    }
    // tail: kk = 260 (k = 260..263); lanes with hi=1 hit the zero pad
    {
        const int k0 = (MIXKP - 4) + hi * 2;   // 260 or 262
        v2f a0, a1, b;
        a0.x = mA0[k0]; a0.y = mA0[k0 + 1];
        a1.x = mA1[k0]; a1.y = mA1[k0 + 1];
        if (k0 < MIXK) {
            const float* wp = W + (size_t)k0 * HD + colbase + lx;
            b.x = wp[0];
            b.y = wp[HD];
        } else {
            b.x = 0.f; b.y = 0.f;
        }
        c0 = __builtin_amdgcn_wmma_f32_16x16x4_f32(false, a0, false, b,
                                                   (short)0, c0, false, false);
        c1 = __builtin_amdgcn_wmma_f32_16x16x4_f32(false, a1, false, b,
                                                   (short)0, c1, false, false);
    }

    // ---- epilogue: bias + SiLU, non-temporal store (write-once output) ----
    const int ncol = colbase + lx;
    const float bv = bias[ncol];
    #pragma unroll
    for (int r = 0; r < 8; ++r) {
        const int row = r + hi * 8;     // D layout: VGPR r -> row r (+8 for hi lanes)
        float x0 = c0[r] + bv;
        __builtin_nontemporal_store(x0 / (1.0f + __expf(-x0)),
                                    out + (size_t)(base + row) * HD + ncol);
        float x1 = c1[r] + bv;
        __builtin_nontemporal_store(x1 / (1.0f + __expf(-x1)),
                                    out + (size_t)(base + 16 + row) * HD + ncol);
    }
}

extern "C" void kernel_launch(void* const* d_in, const int* in_sizes, int n_in,
                              void* d_out, int out_size, void* d_ws, size_t ws_size,
                              hipStream_t stream) {
    const float* feat   = (const float*)d_in[0];
    const float* coords = (const float*)d_in[1];
    const int*   knn    = (const int*)  d_in[2];
    const float* W      = (const float*)d_in[3];
    const float* bias   = (const float*)d_in[4];
    float* out = (float*)d_out;

    const int n = in_sizes[0] / HD;                // 100000
    const int grid = (n + ROWS - 1) / ROWS;        // 3125 full tiles (n % 32 == 0)
    const size_t shmem = (size_t)ROWS * MSTRIDE * sizeof(float)
                       + (size_t)ROWS * KNN * sizeof(int);

    graphconv_wmma_kernel<<<grid, 256, shmem, stream>>>(feat, coords, knn, W, bias, out);
}